// Loss_52175262712573
// MI455X (gfx1250) — compile-verified
//
#include <hip/hip_runtime.h>
#include <hip/hip_bf16.h>

// YOLO-v1 loss for MI455X (gfx1250, wave32).
// Bandwidth-bound streaming kernel: stage AoS tiles into LDS via async
// global->LDS loads (ASYNCcnt path), compute per-cell loss from LDS,
// deterministic two-level tree reduction (graph-capture / replay safe).

namespace {

constexpr int   kS      = 14;
constexpr int   kNb     = 4096;
constexpr int   kNch    = 30;                 // 5*B + C
constexpr int   kCells  = kNb * kS * kS;      // 802816
constexpr int   kBlock  = 256;                // 8 wave32s
constexpr int   kTileF  = kBlock * kNch;      // 7680 floats per tensor tile
constexpr int   kTileV4 = kTileF / 4;         // 1920 float4 per tensor tile
constexpr int   kBlocks = kCells / kBlock;    // 3136 (exact)
constexpr float kInvS   = 1.0f / (float)kS;

// Exact pointee types the async-LDS builtins expect (from clang diagnostic):
// b128 variant takes (int4 addrspace(1)*, int4 addrspace(3)*, imm, imm).
typedef int v4i __attribute__((vector_size(16)));
typedef __attribute__((address_space(1))) v4i* global_v4i_ptr;
typedef __attribute__((address_space(3))) v4i* lds_v4i_ptr;

__device__ __forceinline__ float waveReduceAdd(float v) {
#pragma unroll
  for (int off = 16; off > 0; off >>= 1) v += __shfl_xor(v, off, 32);
  return v;
}

} // namespace

__global__ __launch_bounds__(kBlock) void yolo_loss_partial(
    const float* __restrict__ pred, const float* __restrict__ targ,
    float* __restrict__ part) {
  __shared__ __align__(16) float sP[kTileF];   // 30720 B
  __shared__ __align__(16) float sT[kTileF];   // 30720 B
  __shared__ float sRed[kBlock / 32];

  const int    tid  = threadIdx.x;
  const size_t base = (size_t)blockIdx.x * (size_t)kTileF;

  // ---- Stage both tiles into LDS (coalesced B128 per lane, async DMA) -----
#pragma unroll
  for (int j = 0; j < (kTileV4 + kBlock - 1) / kBlock; ++j) {
    const int idx = j * kBlock + tid;
    if (idx < kTileV4) {
      const float* gp = pred + base + 4 * (size_t)idx;
      const float* gt = targ + base + 4 * (size_t)idx;
      float* lp = &sP[4 * idx];
      float* lt = &sT[4 * idx];
#if defined(__HIP_DEVICE_COMPILE__) && \
    __has_builtin(__builtin_amdgcn_global_load_async_to_lds_b128)
      __builtin_amdgcn_global_load_async_to_lds_b128(
          (global_v4i_ptr)gp, (lds_v4i_ptr)lp, 0, 0);
      __builtin_amdgcn_global_load_async_to_lds_b128(
          (global_v4i_ptr)gt, (lds_v4i_ptr)lt, 0, 0);
#else
      *(float4*)lp = *(const float4*)gp;
      *(float4*)lt = *(const float4*)gt;
#endif
    }
  }
#if defined(__HIP_DEVICE_COMPILE__) && \
    __has_builtin(__builtin_amdgcn_global_load_async_to_lds_b128)
#if __has_builtin(__builtin_amdgcn_s_wait_asynccnt)
  __builtin_amdgcn_s_wait_asynccnt(0);
#else
  asm volatile("s_wait_asynccnt 0" ::: "memory");
#endif
#endif
  __syncthreads();

  // ---- Per-cell loss (reads are LDS bank-conflict free: stride 30) --------
  const float* p = &sP[tid * kNch];
  const float* t = &sT[tid * kNch];

  float cell;
  const float tconf = t[4];  // binarized 0.0 / 1.0 (both conf slots equal)
  if (tconf == 0.0f) {
    // no-object: 0.5 * sum over both box slots of (p_conf - 0)^2
    const float d0 = p[4];
    const float d1 = p[9];
    cell = 0.5f * (d0 * d0 + d1 * d1);
  } else {
    // target box 0 -> corners
    float tx0, ty0, tx1, ty1;
    {
      const float cx = t[0] * kInvS, cy = t[1] * kInvS;
      const float hw = 0.5f * t[2], hh = 0.5f * t[3];
      tx0 = cx - hw; ty0 = cy - hh; tx1 = cx + hw; ty1 = cy + hh;
    }
    const float area_t = (tx1 - tx0) * (ty1 - ty0);

    float iou0 = 0.0f, iou1 = 0.0f;
#pragma unroll
    for (int b = 0; b < 2; ++b) {
      const float* pb = p + 5 * b;
      const float cx = pb[0] * kInvS, cy = pb[1] * kInvS;
      const float hw = 0.5f * pb[2], hh = 0.5f * pb[3];
      const float px0 = cx - hw, py0 = cy - hh;
      const float px1 = cx + hw, py1 = cy + hh;
      const float lx = fmaxf(px0, tx0), ly = fmaxf(py0, ty0);
      const float rx = fminf(px1, tx1), ry = fminf(py1, ty1);
      const float iw = fmaxf(rx - lx, 0.0f), ih = fmaxf(ry - ly, 0.0f);
      const float inter  = iw * ih;
      const float area_p = (px1 - px0) * (py1 - py0);
      const float v = inter / (area_p + area_t - inter);
      if (b == 0) iou0 = v; else iou1 = v;
    }
    // jnp.argmax ties -> first index (box 0)
    const int   r    = (iou1 > iou0) ? 1 : 0;
    const float miou = fmaxf(iou0, iou1);
    const float* pr = p + 5 * r;
    const float* tr = t + 5 * r;

    const float dx = pr[0] - tr[0];
    const float dy = pr[1] - tr[1];
    const float dw = sqrtf(pr[2]) - sqrtf(tr[2]);
    const float dh = sqrtf(pr[3]) - sqrtf(tr[3]);
    const float dc = pr[4] - miou;

    float cls = 0.0f;
#pragma unroll
    for (int c = 10; c < 30; ++c) {
      const float d = p[c] - t[c];
      cls += d * d;
    }
    cell = 5.0f * (dx * dx + dy * dy + dw * dw + dh * dh) + dc * dc + cls;
  }

  // ---- Deterministic block reduction --------------------------------------
  float wsum = waveReduceAdd(cell);
  const int lane = tid & 31;
  const int wid  = tid >> 5;
  if (lane == 0) sRed[wid] = wsum;
  __syncthreads();
  if (wid == 0) {
    float v = (lane < (kBlock / 32)) ? sRed[lane] : 0.0f;
    v = waveReduceAdd(v);
    if (lane == 0) part[blockIdx.x] = v;
  }
}

__global__ __launch_bounds__(256) void yolo_loss_final(
    const float* __restrict__ part, float* __restrict__ out, int n) {
  __shared__ float sRed[8];
  float v = 0.0f;
  for (int i = threadIdx.x; i < n; i += 256) v += part[i];  // fixed order
  v = waveReduceAdd(v);
  const int lane = threadIdx.x & 31;
  const int wid  = threadIdx.x >> 5;
  if (lane == 0) sRed[wid] = v;
  __syncthreads();
  if (wid == 0) {
    float w = (lane < 8) ? sRed[lane] : 0.0f;
    w = waveReduceAdd(w);
    if (lane == 0) out[0] = w * (1.0f / (float)kNb);
  }
}

extern "C" void kernel_launch(void* const* d_in, const int* in_sizes, int n_in,
                              void* d_out, int out_size, void* d_ws, size_t ws_size,
                              hipStream_t stream) {
  (void)in_sizes; (void)n_in; (void)out_size; (void)ws_size;
  const float* pred = (const float*)d_in[0];
  const float* targ = (const float*)d_in[1];
  float* out  = (float*)d_out;
  float* part = (float*)d_ws;  // needs kBlocks * 4 B = 12.5 KB scratch

  yolo_loss_partial<<<kBlocks, kBlock, 0, stream>>>(pred, targ, part);
  yolo_loss_final<<<1, 256, 0, stream>>>(part, out, kBlocks);
}